// Net2_3899830305165
// MI455X (gfx1250) — compile-verified
//
#include <hip/hip_runtime.h>
#include <hip/hip_bf16.h>
#include <math.h>

typedef __attribute__((ext_vector_type(2))) float v2f;
typedef __attribute__((ext_vector_type(8))) float v8f;

#define N_NODES 100000
#define N_EDGES 1600000
#define F_IN    128
#define HID     64
#define N_CLS   40

// ---------------------------------------------------------------------------
// new_w0 = relu(w0 @ E_meta + b0)   [128 x 128], rank-1 + bias, elementwise
// ---------------------------------------------------------------------------
__global__ void k_neww0(const float* __restrict__ w0, const float* __restrict__ Em,
                        const float* __restrict__ b0, float* __restrict__ out) {
  int i = blockIdx.x * blockDim.x + threadIdx.x;
  if (i < F_IN * F_IN) {
    int r = i >> 7, c = i & 127;
    float v = fmaf(w0[r], Em[c], b0[i]);
    out[i] = v > 0.f ? v : 0.f;
  }
}

// ---------------------------------------------------------------------------
// f32 WMMA GEMM: C[M,N] = A[M,K] * B[K,N], row-major, lda==K (contiguous row
// tiles), M%16==0, N%16==0, K%4==0.
// grid.x = M/16 ; blockDim.x = 32*(N/16).  Block stages its 16xK A-tile in
// LDS (coalesced contiguous copy, +1 padding vs 64-bank LDS), each wave owns
// one 16-wide N-tile and runs the K loop with V_WMMA_F32_16X16X4_F32.
// A-frag layout (ISA 7.12.2, 32-bit A 16x4): lanes0-15 M=lane {K=0,1},
// lanes16-31 M=lane-16 {K=2,3}.  B/C/D: rows striped across lanes per VGPR.
// ---------------------------------------------------------------------------
__global__ void k_gemm_f32_wmma(const float* __restrict__ A,
                                const float* __restrict__ B,
                                float* __restrict__ C, int N, int K) {
  extern __shared__ float lAs[];            // 16 * (K+1) floats
  const int tid  = threadIdx.x;
  const int T    = blockDim.x;
  const int m0   = blockIdx.x * 16;

  // cooperative coalesced copy of contiguous 16xK tile
  const float* Ag = A + (size_t)m0 * (size_t)K;
  for (int idx = tid; idx < 16 * K; idx += T) {
    int r = idx / K, c = idx - r * K;
    lAs[r * (K + 1) + c] = Ag[idx];
  }
  __syncthreads();

  const int lane  = tid & 31;
  const int wid   = tid >> 5;
  const int n0    = wid * 16;
  const int mrow  = lane & 15;
  const int khalf = (lane >> 4) * 2;        // 0 for lanes 0-15, 2 for 16-31
  const int ncol  = lane & 15;

  v8f acc = {};
  const float* lA = lAs + mrow * (K + 1) + khalf;
  for (int k = 0; k < K; k += 4) {
    v2f a, b;
    a[0] = lA[k];
    a[1] = lA[k + 1];
    const float* Bp = B + (size_t)(k + khalf) * N + n0 + ncol;
    b[0] = Bp[0];
    b[1] = Bp[N];
    acc = __builtin_amdgcn_wmma_f32_16x16x4_f32(
        /*neg_a=*/false, a, /*neg_b=*/false, b,
        /*c_mod=*/(short)0, acc, /*reuse_a=*/false, /*reuse_b=*/false);
  }

  // C/D layout: VGPR r -> M = r (lanes 0-15) / r+8 (lanes 16-31)
  const int rbase = m0 + ((lane >> 4) << 3);
  float* Cp = C + (size_t)rbase * N + n0 + ncol;
#pragma unroll
  for (int r = 0; r < 8; ++r) Cp[(size_t)r * N] = acc[r];
}

// ---------------------------------------------------------------------------
// degree / dinv
// ---------------------------------------------------------------------------
__global__ void k_zero(float* __restrict__ p, int n) {
  int i = blockIdx.x * blockDim.x + threadIdx.x;
  if (i < n) p[i] = 0.f;
}

__global__ void k_deg(const int* __restrict__ dst, float* __restrict__ deg) {
  int e = blockIdx.x * blockDim.x + threadIdx.x;
  if (e < N_EDGES) atomicAdd(&deg[dst[e]], 1.0f);
}

__global__ void k_dinv(float* __restrict__ d, int n) {
  int i = blockIdx.x * blockDim.x + threadIdx.x;
  if (i < n) d[i] = rsqrtf(1.0f + d[i]);
}

// ---------------------------------------------------------------------------
// agg[i] = dinv[node]^2 * t[i]   (self-loop message; also zero-inits acc)
// ---------------------------------------------------------------------------
__global__ void k_selfinit(const float* __restrict__ t, const float* __restrict__ dinv,
                           float* __restrict__ agg, int n) {
  int i = blockIdx.x * blockDim.x + threadIdx.x;
  if (i < n) {
    float di = dinv[i >> 6];                // HID == 64
    agg[i] = di * di * t[i];
  }
}

// ---------------------------------------------------------------------------
// edge scatter: agg[dst] += dinv[src]*dinv[dst] * t[src]  (64 feats / edge)
// Gather is float4 from the 25.6MB L2-resident table; scatter via f32 atomics.
// ---------------------------------------------------------------------------
__global__ void k_scatter(const int* __restrict__ src, const int* __restrict__ dst,
                          const float* __restrict__ dinv, const float* __restrict__ t,
                          float* __restrict__ agg) {
  int e = blockIdx.x * blockDim.x + threadIdx.x;
  if (e >= N_EDGES) return;
  int s = src[e], d = dst[e];
  float coef = dinv[s] * dinv[d];
  const float4* tp = (const float4*)(t + (size_t)s * HID);
  float* ap = agg + (size_t)d * HID;
#pragma unroll
  for (int q = 0; q < HID / 4; ++q) {
    float4 v = tp[q];
    atomicAdd(ap + 4 * q + 0, coef * v.x);
    atomicAdd(ap + 4 * q + 1, coef * v.y);
    atomicAdd(ap + 4 * q + 2, coef * v.z);
    atomicAdd(ap + 4 * q + 3, coef * v.w);
  }
}

// ---------------------------------------------------------------------------
// h = relu(agg + bias)  (in place)
// ---------------------------------------------------------------------------
__global__ void k_bias_relu(float* __restrict__ h, const float* __restrict__ b, int n) {
  int i = blockIdx.x * blockDim.x + threadIdx.x;
  if (i < n) {
    float v = h[i] + b[i & (HID - 1)];
    h[i] = v > 0.f ? v : 0.f;
  }
}

// ---------------------------------------------------------------------------
// fused classifier: logits = h @ Wl + bl, then log_softmax per node.
// Wl (64x40) + bl cached in LDS; one thread per node (N=40 not WMMA-tileable,
// and this GEMM is 0.5 GFLOP -> noise).
// ---------------------------------------------------------------------------
__global__ void k_logits(const float* __restrict__ h, const float* __restrict__ Wl,
                         const float* __restrict__ bl, float* __restrict__ out) {
  __shared__ float sW[HID * N_CLS];
  __shared__ float sb[N_CLS];
  int tid = threadIdx.x;
  for (int i = tid; i < HID * N_CLS; i += blockDim.x) sW[i] = Wl[i];
  if (tid < N_CLS) sb[tid] = bl[tid];
  __syncthreads();

  int node = blockIdx.x * blockDim.x + tid;
  if (node >= N_NODES) return;

  float acc[N_CLS];
#pragma unroll
  for (int c = 0; c < N_CLS; ++c) acc[c] = sb[c];
  const float* hp = h + (size_t)node * HID;
  for (int k = 0; k < HID; ++k) {
    float hv = hp[k];
#pragma unroll
    for (int c = 0; c < N_CLS; ++c) acc[c] = fmaf(hv, sW[k * N_CLS + c], acc[c]);
  }
  float m = acc[0];
#pragma unroll
  for (int c = 1; c < N_CLS; ++c) m = fmaxf(m, acc[c]);
  float ssum = 0.f;
#pragma unroll
  for (int c = 0; c < N_CLS; ++c) ssum += __expf(acc[c] - m);
  float lse = m + __logf(ssum);
  float* op = out + (size_t)node * N_CLS;
#pragma unroll
  for (int c = 0; c < N_CLS; ++c) op[c] = acc[c] - lse;
}

// ---------------------------------------------------------------------------
extern "C" void kernel_launch(void* const* d_in, const int* in_sizes, int n_in,
                              void* d_out, int out_size, void* d_ws, size_t ws_size,
                              hipStream_t stream) {
  const float* x    = (const float*)d_in[0];   // [100000,128]
  const int*   ei   = (const int*)  d_in[1];   // [2,1600000]
  const float* Em   = (const float*)d_in[2];   // [1,128]
  const float* w0   = (const float*)d_in[3];   // [128,1]
  const float* b0   = (const float*)d_in[4];   // [128,128]
  const float* W1   = (const float*)d_in[5];   // [128,64]
  const float* bc1  = (const float*)d_in[6];   // [64]
  const float* W2   = (const float*)d_in[7];   // [64,64]
  const float* bc2  = (const float*)d_in[8];   // [64]
  const float* Wl   = (const float*)d_in[9];   // [64,40]
  const float* bl   = (const float*)d_in[10];  // [40]
  float* out = (float*)d_out;

  const int* src = ei;
  const int* dst = ei + N_EDGES;

  // workspace layout (floats)
  float* ws = (float*)d_ws;
  float* neww0 = ws;                                   // 16384
  float* dinv  = ws + 16384;                           // 100000 (deg, then dinv)
  float* bufA  = ws + 116384;                          // 12.8M  (h0, later agg2/h2)
  float* bufB  = bufA + (size_t)N_NODES * F_IN;        // 6.4M   (t1, later t2)
  float* bufC  = bufB + (size_t)N_NODES * HID;         // 6.4M   (agg1/h1)

  const int NH = N_NODES * HID;                        // 6,400,000
  dim3 b256(256);

  // 1) new_w0
  k_neww0<<<dim3((F_IN * F_IN + 255) / 256), b256, 0, stream>>>(w0, Em, b0, neww0);

  // 2) h0 = x @ new_w0   (M=100000, N=128, K=128) -> bufA
  k_gemm_f32_wmma<<<dim3(N_NODES / 16), dim3(32 * (F_IN / 16)),
                    16 * (F_IN + 1) * sizeof(float), stream>>>(x, neww0, bufA, F_IN, F_IN);

  // 3) degree -> dinv (shared by both convs)
  k_zero<<<dim3((N_NODES + 255) / 256), b256, 0, stream>>>(dinv, N_NODES);
  k_deg<<<dim3((N_EDGES + 255) / 256), b256, 0, stream>>>(dst, dinv);
  k_dinv<<<dim3((N_NODES + 255) / 256), b256, 0, stream>>>(dinv, N_NODES);

  // 4) conv1: t1 = h0 @ W1 (N=64, K=128) -> bufB ; aggregate -> bufC ; relu
  k_gemm_f32_wmma<<<dim3(N_NODES / 16), dim3(32 * (HID / 16)),
                    16 * (F_IN + 1) * sizeof(float), stream>>>(bufA, W1, bufB, HID, F_IN);
  k_selfinit<<<dim3((NH + 255) / 256), b256, 0, stream>>>(bufB, dinv, bufC, NH);
  k_scatter<<<dim3((N_EDGES + 255) / 256), b256, 0, stream>>>(src, dst, dinv, bufB, bufC);
  k_bias_relu<<<dim3((NH + 255) / 256), b256, 0, stream>>>(bufC, bc1, NH);

  // 5) conv2: t2 = h1 @ W2 (N=64, K=64) -> bufB ; aggregate -> bufA ; relu
  k_gemm_f32_wmma<<<dim3(N_NODES / 16), dim3(32 * (HID / 16)),
                    16 * (HID + 1) * sizeof(float), stream>>>(bufC, W2, bufB, HID, HID);
  k_selfinit<<<dim3((NH + 255) / 256), b256, 0, stream>>>(bufB, dinv, bufA, NH);
  k_scatter<<<dim3((N_EDGES + 255) / 256), b256, 0, stream>>>(src, dst, dinv, bufB, bufA);
  k_bias_relu<<<dim3((NH + 255) / 256), b256, 0, stream>>>(bufA, bc2, NH);

  // 6) logits + log_softmax -> d_out
  k_logits<<<dim3((N_NODES + 255) / 256), b256, 0, stream>>>(bufA, Wl, bl, out);
}